// RetraceLossRecursive_41532333752656
// MI455X (gfx1250) — compile-verified
//
#include <hip/hip_runtime.h>
#include <cstdint>

// Retrace loss on MI455X: HBM-bound (≈1.15 GB traffic -> ~50 us @ 23.3 TB/s).
// No matmul structure -> WMMA inapplicable. CDNA5 features used:
//  - async global->LDS copy engine (ASYNCcnt) for the per-row olog broadcast
//  - non-temporal (TH_LOAD_NT) hints on the read-once streams so the small
//    reusable carry buffers (Ac/Bc/Sc, ~3 MB) stay resident in the 192 MB L2
//  - wave32-aware coalescing: consecutive lanes = consecutive column groups.

typedef float v4f __attribute__((ext_vector_type(4)));

#define GAMMA_F 0.99f
constexpr int T1  = 8192;
constexpr int K   = 4096;
constexpr int N   = T1 - 1;            // 8191 rows of retrace / q
constexpr int TC  = 128;               // rows per chunk
constexpr int C   = (N + TC - 1) / TC; // 64 chunks
constexpr int BLK = 256;
constexpr int V   = 4;                 // columns per thread (b128 streams)
constexpr int CPB = BLK * V;           // 1024 columns per block
constexpr float SCALE = 1.0f / (8191.0f * 4096.0f);

__device__ __forceinline__ v4f ntload4(const float* p) {
  return __builtin_nontemporal_load((const v4f*)p);
}

// Stage 132 consecutive olog values (rows lo .. lo+131, clamped in-bounds)
// into LDS via the CDNA5 async copy engine. 33 lanes x b128 = 132 floats.
// Global base is 16B aligned (lo is a multiple of 128); LDS offsets are
// 16B aligned. Lanes that would run past T1 clamp to T1-4; the LDS slots
// they fill (>=128) are only ever read for non-clamping chunks.
__device__ __forceinline__ void stage_olog_async(float* sOlog,
                                                 const float* olog, int lo) {
  if (threadIdx.x < 33) {
    int base = lo + (int)(threadIdx.x << 2);
    if (base > T1 - 4) base = T1 - 4;
    uint64_t gaddr  = (uint64_t)(uintptr_t)(olog + base);
    uint32_t ldsoff = (uint32_t)(uintptr_t)(sOlog + (threadIdx.x << 2));
    asm volatile("global_load_async_to_lds_b128 %0, %1, off"
                 :: "v"(ldsoff), "v"(gaddr) : "memory");
    asm volatile("s_wait_asynccnt 0x0" ::: "memory");
  }
  __syncthreads();
}

__device__ __forceinline__ float smooth_l1(float d) {
  float ad = fabsf(d);
  return (ad < 1.0f) ? 0.5f * d * d : (ad - 0.5f);
}

// Per-row coefficients for 4 columns: a = r + g*(v - iw*qt), b = g*iw,
// iw = exp(min(tlog - olog, 0)).
__device__ __forceinline__ void ab_row(const v4f tl, const v4f tv,
                                       const v4f ts, const v4f rw,
                                       float so, v4f& a, v4f& b) {
  #pragma unroll
  for (int u = 0; u < V; ++u) {
    const float liw = tl[u] - so;
    const float iw  = __expf(fminf(liw, 0.0f));
    b[u] = GAMMA_F * iw;
    a[u] = fmaf(GAMMA_F, tv[u] - iw * ts[u], rw[u]);
  }
}

// Phase 1: per (chunk, column-group) backward affine composition over TC rows.
__global__ __launch_bounds__(BLK) void retrace_phase1(
    const float* __restrict__ rew,  const float* __restrict__ tsv,
    const float* __restrict__ texp, const float* __restrict__ tlog,
    const float* __restrict__ olog,
    float* __restrict__ Ac, float* __restrict__ Bc) {
  __shared__ float sOlog[136];
  const int c  = blockIdx.y;
  const int lo = c * TC;
  const int hi = min(lo + TC, N) - 1;
  stage_olog_async(sOlog, olog, lo);

  const int k0 = blockIdx.x * CPB + (int)threadIdx.x * V;
  v4f A = {0.0f, 0.0f, 0.0f, 0.0f};
  v4f B = {1.0f, 1.0f, 1.0f, 1.0f};
  int jtop = hi;
  if (hi == N - 1) {                  // terminal row: map is (a = r, b = 0)
    A = ntload4(rew + (size_t)hi * K + k0);
    B = (v4f){0.0f, 0.0f, 0.0f, 0.0f};
    jtop = hi - 1;
  }
  #pragma unroll 2
  for (int j = jtop; j >= lo; --j) {
    const int o0 = j * K + k0;
    const int o1 = o0 + K;            // row j+1
    const v4f tl = ntload4(tlog + o1);
    const v4f tv = ntload4(texp + o1);
    const v4f ts = ntload4(tsv  + o1);
    const v4f rw = ntload4(rew  + o0);
    v4f a, b;
    ab_row(tl, tv, ts, rw, sOlog[j + 1 - lo], a, b);
    #pragma unroll
    for (int u = 0; u < V; ++u) {
      A[u] = fmaf(b[u], A[u], a[u]);
      B[u] *= b[u];
    }
  }
  *(v4f*)(Ac + (size_t)c * K + k0) = A;   // small, reused -> keep cacheable
  *(v4f*)(Bc + (size_t)c * K + k0) = B;
}

// Phase 2: sequential combine over chunks (per column group). S[c] = carry
// INTO chunk c, i.e. ret at the first row of chunk c+1. Ac/Bc are 2 MB and
// should be L2 hits.
__global__ __launch_bounds__(BLK) void retrace_phase2(
    const float* __restrict__ Ac, const float* __restrict__ Bc,
    float* __restrict__ Sc) {
  const int k0 = (blockIdx.x * BLK + (int)threadIdx.x) * V;
  v4f S = {0.0f, 0.0f, 0.0f, 0.0f};
  for (int c = C - 1; c >= 0; --c) {
    const size_t o = (size_t)c * K + k0;
    *(v4f*)(Sc + o) = S;
    const v4f Av = *(const v4f*)(Ac + o);
    const v4f Bv = *(const v4f*)(Bc + o);
    #pragma unroll
    for (int u = 0; u < V; ++u) S[u] = fmaf(Bv[u], S[u], Av[u]);
  }
}

// Phase 3: replay each chunk with its known carry, accumulate smooth-L1.
__global__ __launch_bounds__(BLK) void retrace_phase3(
    const float* __restrict__ q,    const float* __restrict__ rew,
    const float* __restrict__ tsv,  const float* __restrict__ texp,
    const float* __restrict__ tlog, const float* __restrict__ olog,
    const float* __restrict__ Sc,   float* __restrict__ partials) {
  __shared__ float sOlog[136];
  __shared__ float red[BLK];
  const int c  = blockIdx.y;
  const int lo = c * TC;
  const int hi = min(lo + TC, N) - 1;
  stage_olog_async(sOlog, olog, lo);

  const int k0 = blockIdx.x * CPB + (int)threadIdx.x * V;
  v4f x = *(const v4f*)(Sc + (size_t)c * K + k0);  // L2-resident carry
  float lsum = 0.0f;
  int jtop = hi;
  if (hi == N - 1) {
    x = ntload4(rew + (size_t)hi * K + k0);        // ret_{N-1} = r_{N-1}
    const v4f qv = ntload4(q + (size_t)hi * K + k0);
    #pragma unroll
    for (int u = 0; u < V; ++u) lsum += smooth_l1(qv[u] - x[u]);
    jtop = hi - 1;
  }
  #pragma unroll 2
  for (int j = jtop; j >= lo; --j) {
    const int o0 = j * K + k0;
    const int o1 = o0 + K;
    const v4f tl = ntload4(tlog + o1);
    const v4f tv = ntload4(texp + o1);
    const v4f ts = ntload4(tsv  + o1);
    const v4f rw = ntload4(rew  + o0);
    const v4f qv = ntload4(q    + o0);
    v4f a, b;
    ab_row(tl, tv, ts, rw, sOlog[j + 1 - lo], a, b);
    #pragma unroll
    for (int u = 0; u < V; ++u) {
      x[u] = fmaf(b[u], x[u], a[u]);
      lsum += smooth_l1(qv[u] - x[u]);
    }
  }

  // Deterministic block reduction.
  red[threadIdx.x] = lsum;
  __syncthreads();
  for (int off = BLK / 2; off > 0; off >>= 1) {
    if ((int)threadIdx.x < off) red[threadIdx.x] += red[threadIdx.x + off];
    __syncthreads();
  }
  if (threadIdx.x == 0)
    partials[(size_t)blockIdx.y * gridDim.x + blockIdx.x] = red[0];
}

// Final deterministic reduction of block partials -> scalar mean.
__global__ __launch_bounds__(BLK) void retrace_final(
    const float* __restrict__ partials, int n, float* __restrict__ out) {
  __shared__ float red[BLK];
  float s = 0.0f;
  for (int i = threadIdx.x; i < n; i += BLK) s += partials[i];
  red[threadIdx.x] = s;
  __syncthreads();
  for (int off = BLK / 2; off > 0; off >>= 1) {
    if ((int)threadIdx.x < off) red[threadIdx.x] += red[threadIdx.x + off];
    __syncthreads();
  }
  if (threadIdx.x == 0) out[0] = red[0] * SCALE;
}

extern "C" void kernel_launch(void* const* d_in, const int* in_sizes, int n_in,
                              void* d_out, int out_size, void* d_ws,
                              size_t ws_size, hipStream_t stream) {
  (void)in_sizes; (void)n_in; (void)out_size; (void)ws_size;
  const float* q    = (const float*)d_in[0]; // state_trajectory_action_values
  const float* tsv  = (const float*)d_in[1]; // target_state_trajectory_action_values
  const float* texp = (const float*)d_in[2]; // target_expected_state_values
  const float* rew  = (const float*)d_in[3]; // rewards
  const float* olog = (const float*)d_in[4]; // original_log_trajectory_action_probs (T1)
  const float* tlog = (const float*)d_in[5]; // target_log_trajectory_task_action_probs

  float* out = (float*)d_out;
  float* ws  = (float*)d_ws;
  float* Ac = ws;
  float* Bc = Ac + (size_t)C * K;
  float* Sc = Bc + (size_t)C * K;
  float* partials = Sc + (size_t)C * K;

  const dim3 blk(BLK);
  const dim3 grd(K / CPB, C);            // (4, 64) = 256 blocks, 2048 waves

  retrace_phase1<<<grd, blk, 0, stream>>>(rew, tsv, texp, tlog, olog, Ac, Bc);
  retrace_phase2<<<dim3(K / (BLK * V)), blk, 0, stream>>>(Ac, Bc, Sc);
  retrace_phase3<<<grd, blk, 0, stream>>>(q, rew, tsv, texp, tlog, olog, Sc,
                                          partials);
  retrace_final<<<dim3(1), blk, 0, stream>>>(partials, (K / CPB) * C, out);
}